// MoEFeedForward_31722628448795
// MI455X (gfx1250) — compile-verified
//
#include <hip/hip_runtime.h>
#include <math.h>
#include <stdint.h>

// Problem constants (match reference)
#define BATCH 2
#define TSEQ  2048
#define NTOK  (BATCH * TSEQ)   // 4096 tokens
#define DDIM  2048
#define NEXP  16
#define FDIM  1024
#define TOPK  4
#define AUX_COEF 0.001f

// padded LDS row strides (floats): stride % 64 == 4 -> bank 4m+2h, conflict-free b64
#define XS_STRIDE 516   // 512 data + 4 pad
#define HS_STRIDE 132   // 128 data + 4 pad

typedef __attribute__((ext_vector_type(2))) float v2f;
typedef __attribute__((ext_vector_type(8))) float v8f;

// ---------------------------------------------------------------------------
// Workspace layout (4B units):
//   [0..15]               cnt[e]   (int)   tokens routed to expert e
//   [16..31]              S[e]     (float) sum of router probs per expert
//   [32 .. +NEXP*NTOK)    tok list (int)   per-expert token ids (cap NTOK)
//   [.. + NEXP*NTOK)      wt  list (float) per-expert routing weights
// ---------------------------------------------------------------------------

__global__ void moe_zero_kernel(float* out, int n, int* cnt, float* S) {
    int i = blockIdx.x * blockDim.x + threadIdx.x;
    if (i < n) out[i] = 0.0f;
    if (blockIdx.x == 0 && threadIdx.x < NEXP) {
        cnt[threadIdx.x] = 0;
        S[threadIdx.x] = 0.0f;
    }
}

// One wave32 per token: router logits, softmax, top-4, expert-list append.
__global__ void moe_router_kernel(const float* __restrict__ x,
                                  const float* __restrict__ rw,
                                  int* __restrict__ cnt,
                                  float* __restrict__ S,
                                  int* __restrict__ tokl,
                                  float* __restrict__ wtl) {
    const int lane = threadIdx.x & 31;
    const int wave = threadIdx.x >> 5;
    const int t = blockIdx.x * 8 + wave;
    if (t >= NTOK) return;

    const float* xr = x + (size_t)t * DDIM;
    float acc[NEXP];
#pragma unroll
    for (int e = 0; e < NEXP; ++e) acc[e] = 0.0f;

    for (int d = lane; d < DDIM; d += 32) {
        float xv = xr[d];
        const float* w = rw + (size_t)d * NEXP;  // 16 contiguous floats
#pragma unroll
        for (int e = 0; e < NEXP; ++e) acc[e] += xv * w[e];
    }
    // butterfly reduce: every lane ends with the full logit
#pragma unroll
    for (int e = 0; e < NEXP; ++e) {
        float v = acc[e];
        v += __shfl_xor(v, 1);
        v += __shfl_xor(v, 2);
        v += __shfl_xor(v, 4);
        v += __shfl_xor(v, 8);
        v += __shfl_xor(v, 16);
        acc[e] = v;
    }

    if (lane == 0) {
        float mx = acc[0];
#pragma unroll
        for (int e = 1; e < NEXP; ++e) mx = fmaxf(mx, acc[e]);
        float p[NEXP];
        float sum = 0.0f;
#pragma unroll
        for (int e = 0; e < NEXP; ++e) { p[e] = __expf(acc[e] - mx); sum += p[e]; }
        float inv = 1.0f / sum;
#pragma unroll
        for (int e = 0; e < NEXP; ++e) {
            p[e] *= inv;
            atomicAdd(&S[e], p[e]);
        }
        // top-4 by repeated argmax
        int   kid[TOPK];
        float kpv[TOPK];
        unsigned used = 0;
        float wsum = 0.0f;
#pragma unroll
        for (int k = 0; k < TOPK; ++k) {
            int   bi = -1;
            float bv = -1.0f;
#pragma unroll
            for (int e = 0; e < NEXP; ++e) {
                if (!(used & (1u << e)) && p[e] > bv) { bv = p[e]; bi = e; }
            }
            used |= (1u << bi);
            kid[k] = bi; kpv[k] = bv; wsum += bv;
        }
        float denom = 1.0f / fmaxf(wsum, 1e-9f);
#pragma unroll
        for (int k = 0; k < TOPK; ++k) {
            int e = kid[k];
            int pos = atomicAdd(&cnt[e], 1);
            tokl[(size_t)e * NTOK + pos] = t;
            wtl[(size_t)e * NTOK + pos]  = kpv[k] * denom;
        }
    }
}

// Fused expert FFN: 16-token tile per workgroup, 8 waves.
// x tile is staged to LDS in 16x512 chunks via async global->LDS (ASYNCcnt),
// A-fragments come from bank-conflict-free LDS, B-fragments stream from global.
// Phase 1: gate/up WMMA over D, silu*up -> LDS hidden chunk.
// Phase 2: hidden-chunk @ down accumulated in 16 register C-tiles per wave.
__global__ void __launch_bounds__(256)
moe_ffn_kernel(const float* __restrict__ x,
               const float* __restrict__ gatep,
               const float* __restrict__ upp,
               const float* __restrict__ downp,
               const int* __restrict__ cnt,
               const int* __restrict__ tokl,
               const float* __restrict__ wtl,
               float* __restrict__ out) {
    const int e    = blockIdx.y;
    const int tile = blockIdx.x;
    const int ne   = cnt[e];
    const int start = tile * 16;
    if (start >= ne) return;
    const int valid = min(16, ne - start);

    __shared__ __align__(16) float XS[16 * XS_STRIDE];  // x chunk [16 tok][512 d] (padded)
    __shared__ __align__(16) float HS[16 * HS_STRIDE];  // hidden  [16 tok][128 f] (padded)
    __shared__ int   stok[16];
    __shared__ float swt[16];

    if (threadIdx.x < 16) {
        int mm = threadIdx.x;
        if (mm < valid) {
            stok[mm] = tokl[(size_t)e * NTOK + start + mm];
            swt[mm]  = wtl[(size_t)e * NTOK + start + mm];
        } else {
            stok[mm] = 0;
            swt[mm]  = 0.0f;
        }
    }
    __syncthreads();

    const int lane  = threadIdx.x & 31;
    const int wave  = threadIdx.x >> 5;
    const int m     = lane & 15;       // A-matrix row owned by this lane
    const int half  = lane >> 4;
    const int ks    = half * 2;        // k-pair offset per ISA A/B layout
    const int col_n = lane & 15;       // B/C/D column owned by this lane

    const float* Gp = gatep + (size_t)e * DDIM * FDIM;
    const float* Up = upp   + (size_t)e * DDIM * FDIM;
    const float* Dp = downp + (size_t)e * FDIM * DDIM;

    const v8f vzero = {};
    v8f acc[16];
#pragma unroll
    for (int nt = 0; nt < 16; ++nt) acc[nt] = vzero;

    const float* xs_base = XS + m * XS_STRIDE + ks;
    const float* hs_base = HS + m * HS_STRIDE + ks;

    for (int fc = 0; fc < FDIM; fc += 128) {
        // warm the down-proj stream for phase 2 (global_prefetch_b8)
        __builtin_prefetch(Dp + (size_t)fc * DDIM + wave * 256 + col_n, 0, 0);

        const int fcol = fc + wave * 16 + col_n;
        v8f cg = vzero, cu = vzero;

        for (int dc = 0; dc < DDIM; dc += 512) {
            __syncthreads();   // previous XS consumers done
            // async stage: 16 rows x 512 floats -> 2048 b128 copies / 256 threads
            for (int g = threadIdx.x; g < 16 * 128; g += 256) {
                int row = g >> 7;
                int c4  = (g & 127) * 4;
                const float* gsrc = x + (size_t)stok[row] * DDIM + dc + c4;
                uint32_t loff = (uint32_t)(uintptr_t)(XS + row * XS_STRIDE + c4);
                asm volatile("global_load_async_to_lds_b128 %0, %1, off"
                             :: "v"(loff),
                                "v"((unsigned long long)(uintptr_t)gsrc)
                             : "memory");
            }
            asm volatile("s_wait_asynccnt 0x0" ::: "memory");
            __syncthreads();   // staged data visible to all waves

            for (int dd = 0; dd < 512; dd += 4) {
                v2f a = *(const v2f*)(xs_base + dd);     // ds_load_b64, conflict-free
                const size_t dk = (size_t)(dc + dd + ks);
                const float* g0 = Gp + dk * FDIM + fcol;
                const float* u0 = Up + dk * FDIM + fcol;
                v2f bg; bg.x = g0[0]; bg.y = g0[FDIM];
                v2f bu; bu.x = u0[0]; bu.y = u0[FDIM];
                cg = __builtin_amdgcn_wmma_f32_16x16x4_f32(false, a, false, bg,
                                                           (short)0, cg, false, false);
                cu = __builtin_amdgcn_wmma_f32_16x16x4_f32(false, a, false, bu,
                                                           (short)0, cu, false, false);
            }
        }

        // silu(gate) * up -> LDS hidden chunk
#pragma unroll
        for (int r = 0; r < 8; ++r) {
            float g = cg[r];
            float h = (g / (1.0f + __expf(-g))) * cu[r];
            int row = half * 8 + r;              // C/D layout: vgpr r -> row r + 8*half
            HS[row * HS_STRIDE + wave * 16 + col_n] = h;
        }
        __syncthreads();

        // Phase 2: acc[nt] += HS(16x128) @ down[fc..fc+128, wave*256 .. +256)
        for (int kf = 0; kf < 128; kf += 4) {
            v2f a2 = *(const v2f*)(hs_base + kf);        // ds_load_b64, conflict-free
            const float* drow = Dp + (size_t)(fc + kf + ks) * DDIM + wave * 256 + col_n;
#pragma unroll
            for (int nt = 0; nt < 16; ++nt) {
                v2f b2;
                b2.x = drow[nt * 16];
                b2.y = drow[DDIM + nt * 16];
                acc[nt] = __builtin_amdgcn_wmma_f32_16x16x4_f32(false, a2, false, b2,
                                                                (short)0, acc[nt],
                                                                false, false);
            }
        }
        __syncthreads();   // protect HS/XS before next f-chunk
    }

    // Scale by routing weight and scatter-add into merged output.
#pragma unroll
    for (int nt = 0; nt < 16; ++nt) {
        const int col = wave * 256 + nt * 16 + col_n;
#pragma unroll
        for (int r = 0; r < 8; ++r) {
            int row = half * 8 + r;
            if (row < valid) {
                atomicAdd(out + (size_t)stok[row] * DDIM + col, swt[row] * acc[nt][r]);
            }
        }
    }
}

// aux = coef * E/N^2 * sum_e Count_e * S_e   (top-k ids distinct => cnt == Count)
__global__ void moe_aux_kernel(const int* __restrict__ cnt,
                               const float* __restrict__ S,
                               float* __restrict__ aux_out) {
    int lane = threadIdx.x & 31;
    float v = (lane < NEXP) ? (float)cnt[lane] * S[lane] : 0.0f;
    v += __shfl_xor(v, 1);
    v += __shfl_xor(v, 2);
    v += __shfl_xor(v, 4);
    v += __shfl_xor(v, 8);
    v += __shfl_xor(v, 16);
    if (lane == 0) {
        float n2 = (float)NTOK * (float)NTOK;
        aux_out[0] = AUX_COEF * (float)NEXP * v / n2;
    }
}

extern "C" void kernel_launch(void* const* d_in, const int* in_sizes, int n_in,
                              void* d_out, int out_size, void* d_ws, size_t ws_size,
                              hipStream_t stream) {
    const float* x     = (const float*)d_in[0];
    const float* gate  = (const float*)d_in[1];
    const float* up    = (const float*)d_in[2];
    const float* down  = (const float*)d_in[3];
    const float* rw    = (const float*)d_in[4];
    float* out = (float*)d_out;

    int*   cnt  = (int*)d_ws;
    float* S    = (float*)d_ws + 16;
    int*   tokl = (int*)d_ws + 32;
    float* wtl  = (float*)d_ws + 32 + NEXP * NTOK;

    const int ntot = NTOK * DDIM;  // merged output elements (aux written separately)
    moe_zero_kernel<<<(ntot + 255) / 256, 256, 0, stream>>>(out, ntot, cnt, S);

    moe_router_kernel<<<NTOK / 8, 256, 0, stream>>>(x, rw, cnt, S, tokl, wtl);

    dim3 gff(NTOK / 16, NEXP);
    moe_ffn_kernel<<<gff, 256, 0, stream>>>(x, gate, up, down, cnt, tokl, wtl, out);

    moe_aux_kernel<<<1, 32, 0, stream>>>(cnt, S, out + (size_t)NTOK * DDIM);
}